// SparseGraphBuilder_13726715478517
// MI455X (gfx1250) — compile-verified
//
#include <hip/hip_runtime.h>
#include <hip/hip_bf16.h>
#include <math.h>

// ---------------------------------------------------------------------------
// SparseGraphBuilder for MI455X (gfx1250, wave32, WMMA).
//
// d2[i][j] = |pi|^2 + |pj|^2 - 2 pi.pj per 16x16 tile with one
// V_WMMA_F32_16X16X4_F32 via augmentation:
//   A row m  = ( x_m,  y_m,  z_m, 1.0 )          (16x4  f32, K padded 3->4)
//   B col n  = (-2x_n, -2y_n, -2z_n, |p_n|^2 )   (4x16  f32)
//   D = A*B  = |p_n|^2 - 2 p_m.p_n ; then + |p_m|^2 per row.
//
// Fused top-20: BOTH half-waves scan (lane<16: cols 0-7 of row m,
// lane>=16: cols 8-15 of row m), each keeping a register-resident sorted
// top-20; merged once at the end through LDS. Main loop runs with EXEC
// all-1s (WMMA-legal, no per-tile saveexec). Diagonal self-exclusion only
// on the jt==rt tile (uniform scalar branch).
// ---------------------------------------------------------------------------

typedef __attribute__((ext_vector_type(2))) float v2f;
typedef __attribute__((ext_vector_type(8))) float v8f;

#define KNN 20
#define NPTS 4096
#define NBATCH 8
#define NTILES (NPTS / 16)       // 256 column tiles
#define TSTRIDE 21               // LDS row stride: conflict-free for both
                                 // the 2-half-wave writes and reads (mod 64)
#define SELF_EXCLUDE_F 1e10f
#define EPS_F 1e-6f

__global__ void __launch_bounds__(32)
sparse_graph_builder_wmma(const float* __restrict__ pc,    // [B, N, 3]
                          long long* __restrict__ idx_out, // [2, B*N*K] int64
                          float* __restrict__ attrs,       // [B*N*K]
                          float* __restrict__ slocal)      // [B*N]
{
    __shared__ float tile[16 * TSTRIDE]; // 16x16 d2 tile (padded rows)
    __shared__ float sqA[16];            // |p_m|^2 for the 16 A-rows
    __shared__ float kdM[16 * KNN];      // hi half-wave's top-20 (merge)
    __shared__ int   kiM[16 * KNN];

    const int lane = threadIdx.x;        // 0..31, wave32
    const int hi   = lane >> 4;          // operand half / scan half
    const int m    = lane & 15;

    const int b       = blockIdx.x / NTILES;
    const int rt      = blockIdx.x % NTILES;
    const int rowbase = rt * 16;

    const float* __restrict__ P = pc + (size_t)b * NPTS * 3;

    // ---- A operand: every lane loads full (x,y,z) of row m, packs its half.
    const float ax0 = P[(rowbase + m) * 3 + 0];
    const float ay0 = P[(rowbase + m) * 3 + 1];
    const float az0 = P[(rowbase + m) * 3 + 2];
    if (!hi) sqA[m] = ax0 * ax0 + ay0 * ay0 + az0 * az0;

    v2f Av;
    Av[0] = hi ? az0 : ax0;              // K=2 : K=0
    Av[1] = hi ? 1.0f : ay0;             // K=3 : K=1  (augmented 1.0)

    asm volatile("s_wait_dscnt 0" ::: "memory");

    // |p_m|^2 for the 8 rows this lane's C-VGPRs cover (M = r + 8*hi)
    float sqrow[8];
#pragma unroll
    for (int r = 0; r < 8; ++r) sqrow[r] = sqA[r + 8 * hi];

    // ---- register-resident sorted top-K (ascending d2) for this half-row
    float kd[KNN];
    int   ki[KNN];
#pragma unroll
    for (int k = 0; k < KNN; ++k) { kd[k] = 3.0e38f; ki[k] = 0; }

    const int cbase = 8 * hi;            // this lane's scan column base

    for (int jt = 0; jt < NTILES; ++jt) {
        // B operand: lane loads point n = jt*16 + m
        const int n  = jt * 16 + m;
        const float qx = P[n * 3 + 0];
        const float qy = P[n * 3 + 1];
        const float qz = P[n * 3 + 2];
        const float sqn = qx * qx + qy * qy + qz * qz;

        v2f Bv;
        Bv[0] = hi ? (-2.0f * qz) : (-2.0f * qx);   // K=2 : K=0
        Bv[1] = hi ? sqn          : (-2.0f * qy);   // K=3 : K=1  (|p_n|^2)

        if (jt + 1 < NTILES)  // pull next column tile toward WGP$ / L2
            __builtin_prefetch(&P[(jt + 1) * 16 * 3], 0, 3);

        v8f c = {};
        c = __builtin_amdgcn_wmma_f32_16x16x4_f32(
                /*neg_a=*/false, Av, /*neg_b=*/false, Bv,
                /*c_mod=*/(short)0, c, /*reuse_a=*/false, /*reuse_b=*/false);

        // c[r] holds (M = r + 8*hi, Ncol = m): |p_n|^2 - 2 p_m.p_n
        float dv[8];
#pragma unroll
        for (int r = 0; r < 8; ++r)
            dv[r] = fmaxf(c[r] + sqrow[r], 0.0f);

        if (jt == rt) {      // uniform branch: diagonal tile only (1/256)
#pragma unroll
            for (int r = 0; r < 8; ++r)
                if (r + 8 * hi == m) dv[r] += SELF_EXCLUDE_F;
        }

#pragma unroll
        for (int r = 0; r < 8; ++r)
            tile[(r + 8 * hi) * TSTRIDE + m] = dv[r];

        asm volatile("s_wait_dscnt 0" ::: "memory");

        // ALL 32 lanes scan: row m, columns cbase..cbase+7
#pragma unroll 1
        for (int cidx = 0; cidx < 8; ++cidx) {
            const float d = tile[m * TSTRIDE + cbase + cidx];
            if (d < kd[KNN - 1]) {
                kd[KNN - 1] = d;
                ki[KNN - 1] = jt * 16 + cbase + cidx;
#pragma unroll
                for (int s = KNN - 1; s >= 1; --s) {
                    if (kd[s] < kd[s - 1]) {
                        float tf = kd[s]; kd[s] = kd[s - 1]; kd[s - 1] = tf;
                        int   ti = ki[s]; ki[s] = ki[s - 1]; ki[s - 1] = ti;
                    }
                }
            }
        }
        asm volatile("s_wait_dscnt 0" ::: "memory"); // LDS WAR fence
    }

    // ---- merge: hi half-wave publishes its sorted list, lo half merges
    if (hi) {
#pragma unroll
        for (int k = 0; k < KNN; ++k) {
            kdM[m * KNN + k] = kd[k];
            kiM[m * KNN + k] = ki[k];
        }
    }
    asm volatile("s_wait_dscnt 0" ::: "memory");

    if (!hi) {
#pragma unroll 1
        for (int k = 0; k < KNN; ++k) {
            const float d = kdM[m * KNN + k];
            const int   j = kiM[m * KNN + k];
            if (d < kd[KNN - 1]) {
                kd[KNN - 1] = d;
                ki[KNN - 1] = j;
#pragma unroll
                for (int s = KNN - 1; s >= 1; --s) {
                    if (kd[s] < kd[s - 1]) {
                        float tf = kd[s]; kd[s] = kd[s - 1]; kd[s - 1] = tf;
                        int   ti = ki[s]; ki[s] = ki[s - 1]; ki[s - 1] = ti;
                    }
                }
            }
        }

        // ---- epilogue: lane m owns row rowbase+m
        const int i = rowbase + m;
        const long long ibase = (long long)b * NPTS + i;

        const float dk2 = fmaxf(kd[KNN - 1], 0.0f);   // d2 of 20th neighbor
        const float dk  = sqrtf(dk2);                 // d_i
        const float sig = dk + EPS_F;                 // BETA = 1
        const float inv = 1.0f / (2.0f * sig * sig);

        const size_t e0 = ((size_t)b * NPTS + i) * KNN;
        const size_t E  = (size_t)NBATCH * NPTS * KNN;

#pragma unroll
        for (int k = 0; k < KNN; ++k) {
            const float dd = fmaxf(kd[k], 0.0f);      // knn_dist^2
            attrs[e0 + k]        = __expf(-dd * inv);
            idx_out[e0 + k]      = ibase;                        // src
            idx_out[E + e0 + k]  = (long long)b * NPTS + ki[k];  // tgt
        }
        slocal[(size_t)b * NPTS + i] = dk * dk;       // LAMBDA = 1
    }
}

extern "C" void kernel_launch(void* const* d_in, const int* in_sizes, int n_in,
                              void* d_out, int out_size, void* d_ws, size_t ws_size,
                              hipStream_t stream) {
    (void)in_sizes; (void)n_in; (void)out_size; (void)d_ws; (void)ws_size;

    const float* pc = (const float*)d_in[0];          // [8, 4096, 3] f32

    // Output layout (tuple concatenated in return order):
    //   [0]                : edge_indices  int64 [2, E]   (E = B*N*K)
    //   [2*E int64 ...]    : edge_attrs    f32   [E]
    //   [... + E f32]      : s_local       f32   [B*N]
    const size_t E = (size_t)NBATCH * NPTS * KNN;
    long long* idx_out = (long long*)d_out;
    float*     attrs   = (float*)(idx_out + 2 * E);
    float*     slocal  = attrs + E;

    const int nblocks = NBATCH * NTILES;              // 8 * 256 = 2048 waves
    sparse_graph_builder_wmma<<<nblocks, 32, 0, stream>>>(pc, idx_out, attrs, slocal);
}